// EdgeUncertaintyGATLayer_13056700579975
// MI455X (gfx1250) — compile-verified
//
#include <hip/hip_runtime.h>
#include <hip/hip_bf16.h>

// ---------------------------------------------------------------------------
// EdgeUncertaintyGAT layer for MI455X (gfx1250, wave32).
//   B=2, N=1024, IN_F=256, OUT_F=128, META=32, HID=64, k_top=103
// GEMMs use V_WMMA_F32_16X16X4_F32 (exact fp32, matrix pipe) with the B panel
// staged in LDS (8x reuse across the block's waves, conflict-free ds_load_b64
// fragment reads). Edge-score kernel is VALU (ReLU inside the reduction blocks
// GEMM form). Top-k uses a per-row radix-select binary search in LDS.
// ---------------------------------------------------------------------------

typedef float v2f __attribute__((ext_vector_type(2)));
typedef float v8f __attribute__((ext_vector_type(8)));

#define BB     2
#define NN     1024
#define INF    256
#define OUTF   128
#define METAF  32
#define HIDF   64
#define KTOP   103
#define KCHUNK 128   // K staging chunk; all call-site K's are multiples of 128

// ---------------------------------------------------------------------------
// Generic fp32 WMMA GEMM:  C[b] = act(A[b] @ B[b] + bias), row-major.
// 8 waves / block, one 16x16 C tile per wave (128x16 per block).
// B chunk [KCHUNK x 16] staged transposed in LDS: bT[col][k], pad +4 so the
// per-wmma v2f fragment read hits all 64 banks exactly once (banks 4r+2h).
// M multiple of 16, N multiple of 16, K multiple of KCHUNK at all call sites.
// ---------------------------------------------------------------------------
__global__ __launch_bounds__(256) void wmma_gemm_f32(
    const float* __restrict__ A, const float* __restrict__ B,
    float* __restrict__ C, int M, int N, int K,
    long long strideA, long long strideB, long long strideC,
    int applyRelu, const float* __restrict__ bias)
{
    __shared__ float bT[16][KCHUNK + 4];

    const int batch = blockIdx.z;
    A += (size_t)batch * strideA;
    B += (size_t)batch * strideB;
    C += (size_t)batch * strideC;

    const int wave = threadIdx.x >> 5;
    const int lane = threadIdx.x & 31;
    const int row0 = (blockIdx.x * 8 + wave) * 16;   // wave-uniform
    const int col0 = blockIdx.y * 16;                // wave-uniform
    const bool active = (row0 < M);                  // wave-uniform predicate

    const int half = lane >> 4;      // 0: K pair {0,1}, 1: K pair {2,3}
    const int r    = lane & 15;      // row (A) / col (B,C)

    v8f acc = {};
    const float* arow  = A + (size_t)(row0 + r) * K + 2 * half;
    const float* bfrag = &bT[r][2 * half];

    for (int kc = 0; kc < K; kc += KCHUNK) {
        __syncthreads();
        // stage B[kc..kc+KCHUNK) x [col0..col0+16) transposed into LDS
        for (int t = threadIdx.x; t < KCHUNK * 16; t += 256) {
            int kk = t >> 4;         // 0..KCHUNK-1
            int cc = t & 15;         // 0..15
            bT[cc][kk] = B[(size_t)(kc + kk) * N + col0 + cc];
        }
        __syncthreads();

        if (active) {
            if (kc + KCHUNK < K)     // prefetch next A chunk (global_prefetch_b8)
                __builtin_prefetch(arow + kc + KCHUNK, 0, 0);
#pragma unroll 8
            for (int kk = 0; kk < KCHUNK; kk += 4) {
                v2f a = *(const v2f*)(arow + kc + kk);   // global_load_b64
                v2f b = *(const v2f*)(bfrag + kk);       // ds_load_b64
                acc = __builtin_amdgcn_wmma_f32_16x16x4_f32(
                    false, a, false, b, (short)0, acc, false, false);
            }
        }
    }

    if (active) {
#pragma unroll
        for (int i = 0; i < 8; ++i) {
            const int rr = row0 + i + 8 * half;  // C/D layout: VGPR i -> M=i / i+8
            const int cc = col0 + r;
            float v = acc[i];
            if (bias) v += bias[cc];
            if (applyRelu) v = fmaxf(v, 0.0f);
            C[(size_t)rr * N + cc] = v;
        }
    }
}

// ---------------------------------------------------------------------------
// Pack Bcat[128][128] = [W1i | W1j] so hi,hj come from one WMMA GEMM.
// ---------------------------------------------------------------------------
__global__ void pack_bcat_kernel(const float* __restrict__ W1i,
                                 const float* __restrict__ W1j,
                                 float* __restrict__ Bcat)
{
    int idx = blockIdx.x * blockDim.x + threadIdx.x;
    if (idx >= OUTF * 2 * HIDF) return;
    int k = idx >> 7;            // 0..127
    int n = idx & 127;           // 0..127
    Bcat[idx] = (n < HIDF) ? W1i[k * HIDF + n] : W1j[k * HIDF + (n - HIDF)];
}

// ---------------------------------------------------------------------------
// c[b][h] = meta[b] @ W1m + b1   (tiny: 2x64)
// ---------------------------------------------------------------------------
__global__ void hm_kernel(const float* __restrict__ meta,
                          const float* __restrict__ W1m,
                          const float* __restrict__ b1,
                          float* __restrict__ c)
{
    int idx = blockIdx.x * blockDim.x + threadIdx.x;
    if (idx >= BB * HIDF) return;
    int b = idx >> 6, h = idx & 63;
    float s = b1[h];
    for (int m = 0; m < METAF; ++m)
        s = fmaf(meta[b * METAF + m], W1m[m * HIDF + h], s);
    c[idx] = s;
}

// ---------------------------------------------------------------------------
// u[m] = sigmoid(t[m,:] @ U2 + bU2)   -- one wave32 per row, coalesced.
// ---------------------------------------------------------------------------
__global__ void gate_kernel(const float* __restrict__ t,
                            const float* __restrict__ U2,
                            const float* __restrict__ bU2,
                            float* __restrict__ u, int M, int C)
{
    int wave = (blockIdx.x * blockDim.x + threadIdx.x) >> 5;
    int lane = threadIdx.x & 31;
    if (wave >= M) return;
    float acc = 0.0f;
    for (int c = lane; c < C; c += 32)
        acc = fmaf(t[(size_t)wave * C + c], U2[c], acc);
    for (int off = 16; off > 0; off >>= 1)
        acc += __shfl_xor(acc, off, 32);
    if (lane == 0)
        u[wave] = 1.0f / (1.0f + expf(-(acc + bU2[0])));
}

// ---------------------------------------------------------------------------
// eij[b,i,j] = sum_h relu(hi[b,i,h] + hj[b,j,h] + c[b,h]) * W2[h] + b2
// 16x16 edge tile per block; hi/hj tiles staged in LDS (padded vs bank 64).
// ---------------------------------------------------------------------------
__global__ __launch_bounds__(256) void edge_scores_kernel(
    const float* __restrict__ hihj,   // [B*N, 128]: cols 0..63 = hi, 64..127 = hj
    const float* __restrict__ c,      // [B, 64]
    const float* __restrict__ W2,     // [64]
    const float* __restrict__ b2,     // [1]
    float* __restrict__ e)            // [B, N, N]
{
    __shared__ float hiS[16][HIDF + 1];
    __shared__ float hjS[16][HIDF + 1];
    __shared__ float cS[HIDF];
    __shared__ float w2S[HIDF];

    const int b  = blockIdx.z;
    const int i0 = blockIdx.y * 16;
    const int j0 = blockIdx.x * 16;
    const int tid = threadIdx.y * 16 + threadIdx.x;

    for (int t = tid; t < 16 * HIDF; t += 256) {
        int r = t >> 6, h = t & 63;
        hiS[r][h] = hihj[((size_t)(b * NN + i0 + r)) * (2 * HIDF) + h];
        hjS[r][h] = hihj[((size_t)(b * NN + j0 + r)) * (2 * HIDF) + HIDF + h];
    }
    if (tid < HIDF) { cS[tid] = c[b * HIDF + tid]; w2S[tid] = W2[tid]; }
    __syncthreads();

    const int ti = threadIdx.y, tj = threadIdx.x;
    float acc = 0.0f;
#pragma unroll 16
    for (int h = 0; h < HIDF; ++h) {
        float s = hiS[ti][h] + hjS[tj][h] + cS[h];
        acc = fmaf(fmaxf(s, 0.0f), w2S[h], acc);
    }
    e[((size_t)b * NN + (i0 + ti)) * NN + (j0 + tj)] = acc + b2[0];
}

// ---------------------------------------------------------------------------
// Per-row (one block / row): radix-select top-103 threshold, masked softmax
// (non-selected entries contribute exp(0-m)), scale by u[b,j]. In-place.
// ---------------------------------------------------------------------------
__global__ __launch_bounds__(256) void topk_softmax_kernel(
    float* __restrict__ e, const float* __restrict__ u, int kTop)
{
    __shared__ float    vals[NN];
    __shared__ unsigned keys[NN];
    __shared__ float    redf[256];
    __shared__ int      redi[256];

    const int row = blockIdx.x;          // b*N + i
    const int b   = row / NN;
    float* erow   = e + (size_t)row * NN;
    const int tid = threadIdx.x;

    for (int j = tid; j < NN; j += 256) {
        float v = erow[j];
        vals[j] = v;
        unsigned k32 = __float_as_uint(v);               // order-preserving key
        keys[j] = (k32 & 0x80000000u) ? ~k32 : (k32 | 0x80000000u);
    }
    __syncthreads();

    // largest threshold t with count(keys >= t) >= kTop  (32-step binary search)
    unsigned lo = 0u, hi = 0xFFFFFFFFu;
    while (lo < hi) {
        unsigned mid = (unsigned)(((unsigned long long)lo + hi + 1ull) >> 1);
        int cnt = 0;
        for (int j = tid; j < NN; j += 256) cnt += (keys[j] >= mid) ? 1 : 0;
        redi[tid] = cnt;
        __syncthreads();
        for (int s = 128; s > 0; s >>= 1) {
            if (tid < s) redi[tid] += redi[tid + s];
            __syncthreads();
        }
        cnt = redi[0];
        __syncthreads();
        if (cnt >= kTop) lo = mid; else hi = mid - 1;
    }
    const unsigned thr = lo;

    // row max (always a selected element) and selected count
    float mx = -3.402823466e+38f;
    int   sel = 0;
    for (int j = tid; j < NN; j += 256) {
        mx = fmaxf(mx, vals[j]);
        sel += (keys[j] >= thr) ? 1 : 0;
    }
    redf[tid] = mx; redi[tid] = sel;
    __syncthreads();
    for (int s = 128; s > 0; s >>= 1) {
        if (tid < s) {
            redf[tid] = fmaxf(redf[tid], redf[tid + s]);
            redi[tid] += redi[tid + s];
        }
        __syncthreads();
    }
    const float m      = fmaxf(redf[0], 0.0f);   // zeros from masked entries
    const int   numSel = redi[0];
    __syncthreads();

    float ssum = 0.0f;
    for (int j = tid; j < NN; j += 256)
        if (keys[j] >= thr) ssum += expf(vals[j] - m);
    redf[tid] = ssum;
    __syncthreads();
    for (int s = 128; s > 0; s >>= 1) {
        if (tid < s) redf[tid] += redf[tid + s];
        __syncthreads();
    }
    const float zeroTerm = expf(-m);
    const float inv = 1.0f / (redf[0] + (float)(NN - numSel) * zeroTerm);

    for (int j = tid; j < NN; j += 256) {
        float a = (keys[j] >= thr) ? expf(vals[j] - m) : zeroTerm;
        erow[j] = a * inv * u[(size_t)b * NN + j];
    }
}

// ---------------------------------------------------------------------------
extern "C" void kernel_launch(void* const* d_in, const int* in_sizes, int n_in,
                              void* d_out, int out_size, void* d_ws, size_t ws_size,
                              hipStream_t stream)
{
    const float* h    = (const float*)d_in[0];
    const float* meta = (const float*)d_in[1];
    const float* W    = (const float*)d_in[2];
    const float* W1i  = (const float*)d_in[3];
    const float* W1j  = (const float*)d_in[4];
    const float* W1m  = (const float*)d_in[5];
    const float* b1   = (const float*)d_in[6];
    const float* W2   = (const float*)d_in[7];
    const float* b2   = (const float*)d_in[8];
    const float* U1   = (const float*)d_in[9];
    const float* bU1  = (const float*)d_in[10];
    const float* U2   = (const float*)d_in[11];
    const float* bU2  = (const float*)d_in[12];
    float* out = (float*)d_out;

    const int M = BB * NN;   // 2048 node rows

    // workspace layout (floats)
    float* ws   = (float*)d_ws;
    float* Wh   = ws;                          // [2048,128]
    float* hihj = Wh   + (size_t)M * OUTF;     // [2048,128] = [hi|hj]
    float* tbuf = hihj + (size_t)M * OUTF;     // [2048,128] gate hidden
    float* Bcat = tbuf + (size_t)M * OUTF;     // [128,128]  [W1i|W1j]
    float* cbuf = Bcat + OUTF * 2 * HIDF;      // [2,64]  meta@W1m + b1
    float* ubuf = cbuf + BB * HIDF;            // [2048]
    float* eij  = ubuf + M;                    // [2,1024,1024] scores -> att

    // 1) Wh = h @ W                 (WMMA, 2048x128x256)
    {
        dim3 grid((M / 16) / 8, OUTF / 16, 1);
        wmma_gemm_f32<<<grid, 256, 0, stream>>>(h, W, Wh, M, OUTF, INF,
                                                0, 0, 0, 0, nullptr);
    }
    // 2) Bcat = [W1i | W1j]
    pack_bcat_kernel<<<(OUTF * 2 * HIDF + 255) / 256, 256, 0, stream>>>(W1i, W1j, Bcat);
    // 3) [hi|hj] = Wh @ Bcat        (WMMA, 2048x128x128)
    {
        dim3 grid((M / 16) / 8, (2 * HIDF) / 16, 1);
        wmma_gemm_f32<<<grid, 256, 0, stream>>>(Wh, Bcat, hihj, M, 2 * HIDF, OUTF,
                                                0, 0, 0, 0, nullptr);
    }
    // 4) c = meta @ W1m + b1
    hm_kernel<<<1, BB * HIDF, 0, stream>>>(meta, W1m, b1, cbuf);
    // 5) t = relu(h @ U1 + bU1)     (WMMA, 2048x128x256)
    {
        dim3 grid((M / 16) / 8, (INF / 2) / 16, 1);
        wmma_gemm_f32<<<grid, 256, 0, stream>>>(h, U1, tbuf, M, INF / 2, INF,
                                                0, 0, 0, 1, bU1);
    }
    // 6) u = sigmoid(t @ U2 + bU2)
    gate_kernel<<<(M * 32 + 255) / 256, 256, 0, stream>>>(tbuf, U2, bU2, ubuf,
                                                          M, INF / 2);
    // 7) edge scores eij
    {
        dim3 grid(NN / 16, NN / 16, BB);
        dim3 block(16, 16);
        edge_scores_kernel<<<grid, block, 0, stream>>>(hihj, cbuf, W2, b2, eij);
    }
    // 8) top-k + masked softmax + u-gate (in place: eij -> att)
    topk_softmax_kernel<<<BB * NN, 256, 0, stream>>>(eij, ubuf, KTOP);
    // 9) out[b] = att[b] @ Wh[b]    (WMMA batched, 1024x128x1024)
    {
        dim3 grid((NN / 16) / 8, OUTF / 16, BB);
        wmma_gemm_f32<<<grid, 256, 0, stream>>>(
            eij, Wh, out, NN, OUTF, NN,
            (long long)NN * NN, (long long)NN * OUTF, (long long)NN * OUTF,
            0, nullptr);
    }
}